// HouseholderLinear_16922171146574
// MI455X (gfx1250) — compile-verified
//
#include <hip/hip_runtime.h>
#include <math.h>

// fp32 WMMA fragments for V_WMMA_F32_16X16X4_F32 (wave32):
//  A: 16x4  -> v2f/lane  (lanes 0-15: K=ka,ka+1 with ka=0; lanes 16-31: ka=2; row = lane&15)
//  B: 4x16  -> v2f/lane  (same K split; col = lane&15)
//  C/D: 16x16 -> v8f/lane (vgpr e: row = e + 8*(lane>=16); col = lane&15)
typedef __attribute__((ext_vector_type(2))) float v2f;
typedef __attribute__((ext_vector_type(8))) float v8f;

enum { EPI_STORE = 0, EPI_TRI = 1, EPI_IDENT = 2, EPI_BIAS = 3 };

// One lane posts an async 16B memory->LDS copy (tracked by ASYNCcnt).
__device__ __forceinline__ void async_load16(unsigned lds_byte_off, const float* g) {
  asm volatile("global_load_async_to_lds_b128 %0, %1, off"
               :: "v"(lds_byte_off), "v"(g)
               : "memory");
}
__device__ __forceinline__ void wait_async(int n) {
  if (n == 0) asm volatile("s_wait_asynccnt 0" ::: "memory");
  else        asm volatile("s_wait_asynccnt 8" ::: "memory");
}

// C[M x N] = epilogue( alpha * op(A) @ op(B) )
//   TA=0: A is MxK row-major. TA=1: A_mem is KxM row-major (logical A = A_mem^T).
//   TB=0: B is KxN row-major. TB=1: B_mem is NxK row-major (logical B = B_mem^T).
// 64x64 block tile, BK=32, 128 threads = 4 waves (2x2), each wave 2x2 WMMA 16x16 tiles.
// Tiles are staged in RAW global layout via double-buffered async-to-LDS copies;
// fragment indexing adapts to the transpose flags (contiguous b64 reads where possible).
template <int TA, int TB, int EPI>
__global__ __launch_bounds__(128) void gemm_wmma_f32(
    const float* __restrict__ A, const float* __restrict__ B,
    float* __restrict__ C, const float* __restrict__ bias,
    int M, int N, int K, int lda, int ldb, int ldc, float alpha) {
  constexpr int BK = 32;
  // Raw-layout LDS tiles. Row strides 36 / 68 floats: 16B-aligned rows (async chunks
  // stay aligned) and odd-multiple-of-4 dword stride -> conflict-free column access.
  constexpr int ASZ = TA ? (BK * 68) : (64 * 36);
  constexpr int BSZ = TB ? (64 * 36) : (BK * 68);
  __shared__ float AS[2][ASZ];
  __shared__ float BS[2][BSZ];

  const int tid  = threadIdx.x;
  const int lane = tid & 31;
  const int wave = tid >> 5;
  const int wm   = (wave >> 1) * 32;
  const int wn   = (wave & 1) * 32;
  const int hi   = lane >> 4;
  const int l16  = lane & 15;

  const int rowBase = blockIdx.y * 64;
  const int colBase = blockIdx.x * 64;

  auto stage = [&](int buf, int k0) {
    // A tile: 512 16B chunks over 64x32 (or 32x64 when transposed)
#pragma unroll
    for (int q = 0; q < 4; ++q) {
      const int idx = tid + q * 128;
      if (!TA) {
        const int r = idx >> 3, c4 = (idx & 7) << 2;
        async_load16((unsigned)(uintptr_t)&AS[buf][r * 36 + c4],
                     A + (size_t)(rowBase + r) * lda + k0 + c4);
      } else {
        const int r = idx >> 4, c4 = (idx & 15) << 2;
        async_load16((unsigned)(uintptr_t)&AS[buf][r * 68 + c4],
                     A + (size_t)(k0 + r) * lda + rowBase + c4);
      }
    }
#pragma unroll
    for (int q = 0; q < 4; ++q) {
      const int idx = tid + q * 128;
      if (!TB) {
        const int r = idx >> 4, c4 = (idx & 15) << 2;
        async_load16((unsigned)(uintptr_t)&BS[buf][r * 68 + c4],
                     B + (size_t)(k0 + r) * ldb + colBase + c4);
      } else {
        const int r = idx >> 3, c4 = (idx & 7) << 2;
        async_load16((unsigned)(uintptr_t)&BS[buf][r * 36 + c4],
                     B + (size_t)(colBase + r) * ldb + k0 + c4);
      }
    }
  };

  v8f acc[2][2];
#pragma unroll
  for (int i = 0; i < 2; ++i)
#pragma unroll
    for (int j = 0; j < 2; ++j)
#pragma unroll
      for (int e = 0; e < 8; ++e) acc[i][j][e] = 0.0f;

  const int nIter = K / BK;
  stage(0, 0);
  for (int i = 0; i < nIter; ++i) {
    const int buf = i & 1;
    if (i + 1 < nIter) {
      stage(buf ^ 1, (i + 1) * BK);  // prefetch next tile into the other buffer
      wait_async(8);                 // wait only for the current tile's 8 ops
    } else {
      wait_async(0);
    }
    __syncthreads();

#pragma unroll
    for (int kk = 0; kk < BK; kk += 4) {
      const int ka = kk + hi * 2;
      v2f bfrag[2];
#pragma unroll
      for (int tn = 0; tn < 2; ++tn) {
        const int bc = wn + tn * 16 + l16;
        if (!TB) {
          bfrag[tn].x = BS[buf][ka * 68 + bc];
          bfrag[tn].y = BS[buf][(ka + 1) * 68 + bc];
        } else {
          bfrag[tn] = *(const v2f*)&BS[buf][bc * 36 + ka];  // contiguous -> ds_load_b64
        }
      }
#pragma unroll
      for (int tm = 0; tm < 2; ++tm) {
        const int ar = wm + tm * 16 + l16;
        v2f afrag;
        if (!TA) {
          afrag = *(const v2f*)&AS[buf][ar * 36 + ka];      // contiguous -> ds_load_b64
        } else {
          afrag.x = AS[buf][ka * 68 + ar];
          afrag.y = AS[buf][(ka + 1) * 68 + ar];
        }
#pragma unroll
        for (int tn = 0; tn < 2; ++tn)
          acc[tm][tn] = __builtin_amdgcn_wmma_f32_16x16x4_f32(
              false, afrag, false, bfrag[tn], (short)0, acc[tm][tn], false, false);
      }
    }
    __syncthreads();  // all waves done reading buf before it is overwritten
  }

  // ---- epilogue
#pragma unroll
  for (int tm = 0; tm < 2; ++tm)
#pragma unroll
    for (int tn = 0; tn < 2; ++tn) {
      const int cb = colBase + wn + tn * 16 + l16;
#pragma unroll
      for (int e = 0; e < 8; ++e) {
        const int rb = rowBase + wm + tm * 16 + e + hi * 8;
        float v = acc[tm][tn][e];
        float o;
        if (EPI == EPI_STORE)      o = alpha * v;
        else if (EPI == EPI_TRI)   o = (rb < cb) ? 2.0f * v : ((rb == cb) ? 1.0f : 0.0f);
        else if (EPI == EPI_IDENT) o = alpha * v + ((rb == cb) ? 1.0f : 0.0f);
        else                       o = v + bias[cb];
        C[(size_t)rb * ldc + cb] = o;
      }
    }
}

// Row-normalize: V[r,:] = vec[r,:] / ||vec[r,:]||
__global__ __launch_bounds__(256) void normalize_rows(
    const float* __restrict__ vec, float* __restrict__ V, int d) {
  __shared__ float red[256];
  const int row = blockIdx.x;
  const float4* src = (const float4*)(vec + (size_t)row * d);
  float s = 0.0f;
  for (int i = threadIdx.x; i < (d >> 2); i += 256) {
    float4 v = src[i];
    s += v.x * v.x + v.y * v.y + v.z * v.z + v.w * v.w;
  }
  red[threadIdx.x] = s;
  __syncthreads();
  for (int off = 128; off > 0; off >>= 1) {
    if (threadIdx.x < off) red[threadIdx.x] += red[threadIdx.x + off];
    __syncthreads();
  }
  const float invn = 1.0f / sqrtf(red[0]);
  float4* dst = (float4*)(V + (size_t)row * d);
  for (int i = threadIdx.x; i < (d >> 2); i += 256) {
    float4 v = src[i];
    dst[i] = make_float4(v.x * invn, v.y * invn, v.z * invn, v.w * invn);
  }
}

__global__ void fill_zero(float* __restrict__ p, size_t n) {
  size_t i = (size_t)blockIdx.x * blockDim.x + threadIdx.x;
  if (i < n) p[i] = 0.0f;
}

// Invert 64x64 unit-upper-triangular diagonal blocks of Amat into Sinv.
// Row recursion Sinv[i,k] = -sum_{j=i..k-1} Sinv[i,j]*A[j,k]: each thread touches
// only its own Sinv row + read-only A -> no barriers inside the sweep.
__global__ __launch_bounds__(64) void tri_diag_inv(
    const float* __restrict__ Amat, float* __restrict__ Sinv, int m) {
  __shared__ float Ab[64][65];
  __shared__ float Sb[64][65];
  const int b = blockIdx.x * 64;
  const int t = threadIdx.x;
  for (int c = 0; c < 64; ++c) {
    Ab[t][c] = Amat[(size_t)(b + t) * m + b + c];
    Sb[t][c] = (t == c) ? 1.0f : 0.0f;
  }
  __syncthreads();
  for (int k = t + 1; k < 64; ++k) {
    float s = 0.0f;
    for (int j = t; j < k; ++j) s += Sb[t][j] * Ab[j][k];
    Sb[t][k] = -s;
  }
  __syncthreads();
  for (int c = 0; c < 64; ++c)
    Sinv[(size_t)(b + t) * m + b + c] = Sb[t][c];
}

// Block-diagonal t of the triangular inverse:
//   Sinv[ib,ib+t] = -( sum_{kb=ib..ib+t-1} Sinv[ib,kb] @ A[kb,ib+t] ) @ Sinv[ib+t,ib+t]
// Launched for t = 1..nblk-1 sequentially (stream order provides the dependency).
__global__ __launch_bounds__(256) void tri_offdiag(
    const float* __restrict__ Amat, float* __restrict__ Sinv, int m, int t) {
  __shared__ float M1[64][65];
  const int ib = blockIdx.x, jb = ib + t;
  const int r0 = ib * 64, c0 = jb * 64;
  const int L0 = ib * 64, L1 = jb * 64;
  for (int idx = threadIdx.x; idx < 4096; idx += 256) {
    int r = idx >> 6, c = idx & 63;
    float s = 0.0f;
    for (int l = L0; l < L1; ++l)
      s += Sinv[(size_t)(r0 + r) * m + l] * Amat[(size_t)l * m + c0 + c];
    M1[r][c] = s;
  }
  __syncthreads();
  for (int idx = threadIdx.x; idx < 4096; idx += 256) {
    int r = idx >> 6, c = idx & 63;
    float s = 0.0f;
    for (int j = 0; j <= c; ++j)  // diagonal block is upper triangular
      s += M1[r][j] * Sinv[(size_t)(c0 + j) * m + c0 + c];
    Sinv[(size_t)(r0 + r) * m + c0 + c] = -s;
  }
}

extern "C" void kernel_launch(void* const* d_in, const int* in_sizes, int n_in,
                              void* d_out, int out_size, void* d_ws, size_t ws_size,
                              hipStream_t stream) {
  const float* x       = (const float*)d_in[0];
  const float* vectors = (const float*)d_in[1];
  const float* bias    = (const float*)d_in[2];
  float* out = (float*)d_out;

  const int dim = in_sizes[2];        // 1024
  const int m   = in_sizes[1] / dim;  // 1024
  const int N   = in_sizes[0] / dim;  // 8192

  // Workspace: V (m x dim) | bufA (A-matrix, then W) | bufB (Sinv, then Q)
  const size_t szV = (size_t)m * dim;
  const size_t szA = (size_t)m * m > (size_t)m * dim ? (size_t)m * m : (size_t)m * dim;
  float* V    = (float*)d_ws;
  float* bufA = V + szV;
  float* bufB = bufA + szA;

  const int nblk = m / 64;

  // 1) V = row-normalized vectors
  normalize_rows<<<m, 256, 0, stream>>>(vectors, V, dim);

  // 2) bufA = I + 2*strict_upper(V @ V^T)    [TRI epilogue on Gram GEMM]
  gemm_wmma_f32<0, 1, EPI_TRI><<<dim3(m / 64, m / 64), 128, 0, stream>>>(
      V, V, bufA, nullptr, m, m, dim, dim, dim, m, 1.0f);

  // 3) bufB = bufA^{-1} (unit upper triangular inverse), blocked
  {
    size_t n = (size_t)m * m;
    fill_zero<<<(unsigned)((n + 255) / 256), 256, 0, stream>>>(bufB, n);
  }
  tri_diag_inv<<<nblk, 64, 0, stream>>>(bufA, bufB, m);
  for (int t = 1; t < nblk; ++t)
    tri_offdiag<<<nblk - t, 256, 0, stream>>>(bufA, bufB, m, t);

  // 4) bufA := W = 2 * bufB @ V          (m x dim)
  gemm_wmma_f32<0, 0, EPI_STORE><<<dim3(dim / 64, m / 64), 128, 0, stream>>>(
      bufB, V, bufA, nullptr, m, dim, m, m, dim, dim, 2.0f);

  // 5) bufB := Q = I - V^T @ W           (dim x dim)
  gemm_wmma_f32<1, 0, EPI_IDENT><<<dim3(dim / 64, dim / 64), 128, 0, stream>>>(
      V, bufA, bufB, nullptr, dim, dim, m, dim, dim, dim, -1.0f);

  // 6) out = x @ Q + bias                (N x dim) — the only N-sized pass
  gemm_wmma_f32<0, 0, EPI_BIAS><<<dim3(dim / 64, N / 64), 128, 0, stream>>>(
      x, bufB, out, bias, N, dim, dim, dim, dim, dim, 1.0f);
}